// MutualNet_85323820302978
// MI455X (gfx1250) — compile-verified
//
#include <hip/hip_runtime.h>
#include <math.h>

typedef __attribute__((ext_vector_type(2))) float v2f;
typedef __attribute__((ext_vector_type(8))) float v8f;

#define HW   16384
#define CC   256
#define NBAT 8
#define NS   64
#define NK   8

// ---------------- zero workspace region (graph-replay safe) ----------------
__global__ void k_zero(float* __restrict__ p, int n) {
    int i = blockIdx.x * 256 + threadIdx.x;
    if (i < n) p[i] = 0.f;
}

// ---------------- contour mask: softmax over 2 channels, take ch1 ----------
__global__ void k_cm(const float* __restrict__ ct, float* __restrict__ cm) {
    int i = blockIdx.x * 256 + threadIdx.x;       // B*HW threads
    int b = i / HW, n = i % HW;
    float c0 = ct[(size_t)b * 2 * HW + n];
    float c1 = ct[(size_t)b * 2 * HW + HW + n];
    cm[i] = 1.f / (1.f + __expf(c0 - c1));        // == softmax[:,1]
}

// ---------------- x_proj = Wp(64x256) @ x(256xHW) + b  [WMMA f32 16x16x4] --
__global__ __launch_bounds__(256) void k_xproj(const float* __restrict__ x,
                                               const float* __restrict__ wp,
                                               const float* __restrict__ bp,
                                               float* __restrict__ xproj) {
    const int b    = blockIdx.x;
    const int wave = threadIdx.x >> 5;
    const int lane = threadIdx.x & 31;
    const int n0   = blockIdx.y * 128 + wave * 16;
    const int col  = lane & 15;                 // N col within frag / M row for A
    const int kr   = (lane < 16) ? 0 : 2;       // K split across lane halves
    const float* xb = x + (size_t)b * CC * HW;

    v8f acc[4] = {};
    for (int c0 = 0; c0 < CC; c0 += 4) {
        v2f bf;
        bf.x = xb[(c0 + kr)     * HW + n0 + col];
        bf.y = xb[(c0 + kr + 1) * HW + n0 + col];
#pragma unroll
        for (int mt = 0; mt < 4; ++mt) {
            v2f a;
            a.x = wp[(mt * 16 + col) * CC + c0 + kr];
            a.y = wp[(mt * 16 + col) * CC + c0 + kr + 1];
            acc[mt] = __builtin_amdgcn_wmma_f32_16x16x4_f32(
                false, a, false, bf, (short)0, acc[mt], false, false);
        }
    }
    const int rbase = (lane < 16) ? 0 : 8;
    float* outb = xproj + (size_t)b * NS * HW;
#pragma unroll
    for (int mt = 0; mt < 4; ++mt)
#pragma unroll
        for (int r = 0; r < 8; ++r) {
            int rowm = mt * 16 + rbase + r;
            outb[(size_t)rowm * HW + n0 + col] = acc[mt][r] + bp[rowm];
        }
}

// ---------------- adaptive pool(10x10) of x_proj*cm, inner 8x8 -------------
__global__ void k_pool(const float* __restrict__ xproj,
                       const float* __restrict__ cm,
                       float* __restrict__ xa) {
    int b = blockIdx.x, s = blockIdx.y;
    int t = threadIdx.x;                          // 64 threads: (oy,ox)
    int oy = t >> 3, ox = t & 7;
    int sh = ((oy + 1) * 128) / 10, eh = ((oy + 2) * 128 + 9) / 10;
    int sw = ((ox + 1) * 128) / 10, ew = ((ox + 2) * 128 + 9) / 10;
    const float* xp = xproj + (size_t)b * NS * HW + (size_t)s * HW;
    const float* cb = cm + (size_t)b * HW;
    float sum = 0.f;
    for (int h = sh; h < eh; ++h)
        for (int w = sw; w < ew; ++w) {
            int n = h * 128 + w;
            sum += xp[n] * cb[n];
        }
    xa[((size_t)b * NS + s) * 64 + t] = sum / (float)((eh - sh) * (ew - sw));
}

// ------- proj = softmax_m( x_anchor^T(64x64) @ x_proj(64xHW) ) [WMMA] ------
__global__ __launch_bounds__(256) void k_proj(const float* __restrict__ xa,
                                              const float* __restrict__ xproj,
                                              float* __restrict__ projout) {
    const int b    = blockIdx.x;
    const int wave = threadIdx.x >> 5;
    const int lane = threadIdx.x & 31;
    const int n0   = blockIdx.y * 128 + wave * 16;
    const int col  = lane & 15;
    const int kr   = (lane < 16) ? 0 : 2;
    const float* xab = xa + (size_t)b * NS * 64;          // [s][m]
    const float* xpb = xproj + (size_t)b * NS * HW;

    v8f acc[4] = {};
    for (int c0 = 0; c0 < NS; c0 += 4) {
        v2f bf;
        bf.x = xpb[(c0 + kr)     * HW + n0 + col];
        bf.y = xpb[(c0 + kr + 1) * HW + n0 + col];
#pragma unroll
        for (int mt = 0; mt < 4; ++mt) {
            v2f a;                                        // A[m][s] = xa[s][m]
            a.x = xab[(c0 + kr)     * 64 + mt * 16 + col];
            a.y = xab[(c0 + kr + 1) * 64 + mt * 16 + col];
            acc[mt] = __builtin_amdgcn_wmma_f32_16x16x4_f32(
                false, a, false, bf, (short)0, acc[mt], false, false);
        }
    }
    // softmax over m=64: lane holds 32 of the 64 values for its column,
    // partner lane (xor 16) holds the rest.
    float mx = -1e30f;
#pragma unroll
    for (int mt = 0; mt < 4; ++mt)
#pragma unroll
        for (int r = 0; r < 8; ++r) mx = fmaxf(mx, acc[mt][r]);
    mx = fmaxf(mx, __shfl_xor(mx, 16, 32));
    float sum = 0.f;
#pragma unroll
    for (int mt = 0; mt < 4; ++mt)
#pragma unroll
        for (int r = 0; r < 8; ++r) {
            float e = __expf(acc[mt][r] - mx);
            acc[mt][r] = e;
            sum += e;
        }
    sum += __shfl_xor(sum, 16, 32);
    float inv = 1.f / sum;
    const int rbase = (lane < 16) ? 0 : 8;
    float* outb = projout + (size_t)b * NS * HW;
#pragma unroll
    for (int mt = 0; mt < 4; ++mt)
#pragma unroll
        for (int r = 0; r < 8; ++r) {
            int rowm = mt * 16 + rbase + r;
            outb[(size_t)rowm * HW + n0 + col] = acc[mt][r] * inv;
        }
}

// ---------------- GraphNet: d2 -> soft (out) + ssum/wsum accumulation ------
__global__ __launch_bounds__(256) void k_graph(const float* __restrict__ x,
                                               const float* __restrict__ anchor,
                                               const float* __restrict__ sigma_p,
                                               float* __restrict__ softout,
                                               float* __restrict__ wsum,
                                               float* __restrict__ ssum) {
    __shared__ float wk1[NK * CC];   // 1/sigma^2
    __shared__ float wk2[NK * CC];   // 2*a/sigma^2
    __shared__ float ck[NK];         // sum a^2/sigma^2
    __shared__ float ss[NK * 256];   // soft tile
    const int b = blockIdx.x, tile = blockIdx.y;
    const int t = threadIdx.x;
    const int n = tile * 256 + t;

    for (int k = 0; k < NK; ++k) {
        float sg = 1.f / (1.f + __expf(-sigma_p[k * CC + t]));
        float iv = 1.f / (sg * sg);
        wk1[k * CC + t] = iv;
        wk2[k * CC + t] = 2.f * anchor[k * CC + t] * iv;
    }
    __syncthreads();
    if (t < NK) {
        float s = 0.f;
        for (int c = 0; c < CC; ++c) {
            float a = anchor[t * CC + c];
            s += a * a * wk1[t * CC + c];
        }
        ck[t] = s;
    }
    __syncthreads();

    const float* xb = x + (size_t)b * CC * HW;
    float acc1[NK] = {}, acc2[NK] = {};
    for (int c = 0; c < CC; ++c) {
        float xv = xb[(size_t)c * HW + n];       // coalesced; L2-resident
        float x2 = xv * xv;
#pragma unroll
        for (int k = 0; k < NK; ++k) {
            acc1[k] = fmaf(x2, wk1[k * CC + c], acc1[k]);
            acc2[k] = fmaf(xv, wk2[k * CC + c], acc2[k]);
        }
    }
    float logit[NK], mx = -1e30f;
#pragma unroll
    for (int k = 0; k < NK; ++k) {
        logit[k] = -0.5f * (acc1[k] - acc2[k] + ck[k]);
        mx = fmaxf(mx, logit[k]);
    }
    float sum = 0.f;
#pragma unroll
    for (int k = 0; k < NK; ++k) {
        logit[k] = __expf(logit[k] - mx);
        sum += logit[k];
    }
    float inv = 1.f / sum;
#pragma unroll
    for (int k = 0; k < NK; ++k) {
        float sv = logit[k] * inv;
        softout[(size_t)b * NK * HW + (size_t)k * HW + n] = sv;
        ss[k * 256 + t] = sv;
    }
    __syncthreads();
    if (t < NK) {
        float s = 0.f;
        for (int p = 0; p < 256; ++p) s += ss[t * 256 + p];
        atomicAdd(&ssum[b * NK + t], s);
    }
    // pass 2: thread t owns channel c=t; x row stays hot in L2
    float wacc[NK] = {};
    const float* xrow = xb + (size_t)t * HW + tile * 256;
    for (int p = 0; p < 256; ++p) {
        float xv = xrow[p];
#pragma unroll
        for (int k = 0; k < NK; ++k)
            wacc[k] = fmaf(ss[k * 256 + p], xv, wacc[k]);   // broadcast LDS read
    }
#pragma unroll
    for (int k = 0; k < NK; ++k)
        atomicAdd(&wsum[((size_t)b * NK + k) * CC + t], wacc[k]);
}

// ---------------- nodes + normalize + 3x CascadeGCN + relu -----------------
__global__ __launch_bounds__(256) void k_nodes_gcn(const float* __restrict__ wsum,
                                                   const float* __restrict__ ssum,
                                                   const float* __restrict__ anchor,
                                                   const float* __restrict__ sigma_p,
                                                   const float* __restrict__ w1,
                                                   const float* __restrict__ w2,
                                                   const float* __restrict__ w3,
                                                   float* __restrict__ gout) {
    __shared__ float flat[2048];   // nodes flat / G matrix (reinterpreted)
    __shared__ float adjm[64];
    __shared__ float rn[NK];
    __shared__ float red[256];
    const int b = blockIdx.x, t = threadIdx.x;   // t == channel c

    for (int k = 0; k < NK; ++k) {
        float sg = 1.f / (1.f + __expf(-sigma_p[k * CC + t]));
        float sv = ssum[b * NK + k];
        float wv = wsum[((size_t)b * NK + k) * CC + t];
        flat[k * CC + t] = (wv - anchor[k * CC + t] * sv) / sg / (sv + 1e-9f);
    }
    __syncthreads();
    if (t < NK) {
        float s = 0.f;
        for (int c = 0; c < CC; ++c) { float v = flat[t * CC + c]; s += v * v; }
        rn[t] = 1.f / fmaxf(sqrtf(s), 1e-12f);
    }
    __syncthreads();
    float local = 0.f;
    for (int k = 0; k < NK; ++k) {
        float v = flat[k * CC + t] * rn[k];
        flat[k * CC + t] = v;
        local += v * v;
    }
    red[t] = local;
    __syncthreads();
    for (int off = 128; off > 0; off >>= 1) {
        if (t < off) red[t] += red[t + off];
        __syncthreads();
    }
    float fn = 1.f / fmaxf(sqrtf(red[0]), 1e-12f);
    for (int k = 0; k < NK; ++k) flat[k * CC + t] *= fn;
    __syncthreads();

    // g = flat viewed as (C,K): G[c][k] = flat[c*8+k]
    const float* Ws[3] = { w1, w2, w3 };
    for (int it = 0; it < 3; ++it) {
        const float* W = Ws[it];
        if (t < 64) {
            int k = t >> 3, j = t & 7;
            float s = 0.f;
            for (int c = 0; c < CC; ++c) s += flat[c * NK + k] * flat[c * NK + j];
            adjm[t] = s;
        }
        __syncthreads();
        if (t < NK) {
            float m = -1e30f;
            for (int j = 0; j < 8; ++j) m = fmaxf(m, adjm[t * 8 + j]);
            float s = 0.f, e[8];
            for (int j = 0; j < 8; ++j) { e[j] = __expf(adjm[t * 8 + j] - m); s += e[j]; }
            for (int j = 0; j < 8; ++j) adjm[t * 8 + j] = e[j] / s;
        }
        __syncthreads();
        float supp[8] = {};
        for (int c = 0; c < CC; ++c) {
            float wv = W[c * CC + t];                 // coalesced over t
#pragma unroll
            for (int j = 0; j < 8; ++j)
                supp[j] = fmaf(flat[c * NK + j], wv, supp[j]);  // broadcast
        }
        float newv[8];
#pragma unroll
        for (int k = 0; k < 8; ++k) {
            float s = 0.f;
            for (int j = 0; j < 8; ++j) s += adjm[k * 8 + j] * supp[j];
            newv[k] = s;
        }
        __syncthreads();
        for (int k = 0; k < 8; ++k) flat[t * NK + k] = newv[k];
        __syncthreads();
    }
    for (int k = 0; k < 8; ++k)
        gout[((size_t)b * CC + t) * NK + k] = fmaxf(flat[t * NK + k], 0.f);
}

extern "C" void kernel_launch(void* const* d_in, const int* in_sizes, int n_in,
                              void* d_out, int out_size, void* d_ws, size_t ws_size,
                              hipStream_t stream) {
    (void)in_sizes; (void)n_in; (void)out_size; (void)ws_size;
    const float* x   = (const float*)d_in[0];
    const float* ct  = (const float*)d_in[1];
    // d_in[2], d_in[3] (conv_state_w/b) unused by the reference
    const float* wp  = (const float*)d_in[4];
    const float* bp  = (const float*)d_in[5];
    const float* anc = (const float*)d_in[6];
    const float* sgp = (const float*)d_in[7];
    const float* g1  = (const float*)d_in[8];
    const float* g2  = (const float*)d_in[9];
    const float* g3  = (const float*)d_in[10];

    float* out      = (float*)d_out;
    float* g_out    = out;                         // B*C*K   = 16384
    float* proj_out = out + 16384;                 // B*64*HW = 8388608
    float* soft_out = out + 16384 + 8388608;       // B*8*HW  = 1048576

    float* ws    = (float*)d_ws;
    float* cm    = ws;                             // 131072
    float* xproj = ws + 131072;                    // 8388608
    float* xa    = xproj + 8388608;                // 32768
    float* wsum  = xa + 32768;                     // 16384
    // ssum = wsum + 16384 (64 floats), zeroed together with wsum

    k_zero<<<dim3(65), 256, 0, stream>>>(wsum, 16384 + 64);
    k_cm<<<dim3(NBAT * HW / 256), 256, 0, stream>>>(ct, cm);
    k_xproj<<<dim3(NBAT, HW / 128), 256, 0, stream>>>(x, wp, bp, xproj);
    k_pool<<<dim3(NBAT, NS), 64, 0, stream>>>(xproj, cm, xa);
    k_proj<<<dim3(NBAT, HW / 128), 256, 0, stream>>>(xa, xproj, proj_out);
    k_graph<<<dim3(NBAT, HW / 256), 256, 0, stream>>>(x, anc, sgp, soft_out, wsum,
                                                      wsum + 16384);
    k_nodes_gcn<<<dim3(NBAT), 256, 0, stream>>>(wsum, wsum + 16384, anc, sgp,
                                                g1, g2, g3, g_out);
}